// MPNN_83751862272705
// MI455X (gfx1250) — compile-verified
//
#include <hip/hip_runtime.h>

#define NNODES 25000
#define NEDGES 50000
#define NFEAT  32
#define BFEAT  16
#define H      64
#define NG     500

typedef __attribute__((ext_vector_type(2))) float        v2f;
typedef __attribute__((ext_vector_type(8))) float        v8f;
typedef __attribute__((ext_vector_type(4))) unsigned int v4u;
typedef __attribute__((ext_vector_type(8))) int          v8i;
typedef __attribute__((ext_vector_type(4))) int          v4i;

#ifndef __has_builtin
#define __has_builtin(x) 0
#endif
#if __has_builtin(__builtin_amdgcn_tensor_load_to_lds) && \
    __has_builtin(__builtin_amdgcn_s_wait_tensorcnt)
#define USE_TDM 1
#else
#define USE_TDM 0
#endif

// ---------------------------------------------------------------- utilities
__global__ void zero_kernel(float* __restrict__ p, int n) {
    int i = blockIdx.x * blockDim.x + threadIdx.x;
    if (i < n) p[i] = 0.0f;
}

// z[e,h] = relu(ea[e,:] @ w1 + b1)
__global__ void edge_mlp1_kernel(const float* __restrict__ ea,
                                 const float* __restrict__ w1,
                                 const float* __restrict__ b1,
                                 float* __restrict__ z) {
    int idx = blockIdx.x * blockDim.x + threadIdx.x;
    if (idx >= NEDGES * H) return;
    int e = idx >> 6, h = idx & 63;
    const float* row = ea + e * BFEAT;
    float acc = b1[h];
#pragma unroll
    for (int k = 0; k < BFEAT; ++k) acc += row[k] * w1[k * H + h];
    z[idx] = fmaxf(acc, 0.0f);
}

#if USE_TDM
// Issue a TDM DMA of `nelem` contiguous f32 from `gptr` into LDS offset
// `lds_off` (1-D tile). D# layout per CDNA5 ISA ch.8 (group0/group1).
// This toolchain exposes the 6-arg builtin:
//   (uint32x4 g0, int32x8 g1, int32x4 g2, int32x4 g3, int32x8 g4, i32 cpol)
__device__ __forceinline__ void tdm_load_row(const float* gptr, unsigned lds_off,
                                             int nelem) {
    unsigned long long ga = (unsigned long long)(size_t)gptr;
    unsigned dim0 = (unsigned)nelem;
    v4u g0;
    g0[0] = 1u;                                            // count=1, no gather
    g0[1] = lds_off;                                       // lds_addr
    g0[2] = (unsigned)(ga & 0xFFFFFFFFu);                  // global_addr[31:0]
    g0[3] = (unsigned)((ga >> 32) & 0x01FFFFFFu)           // global_addr[56:32]
          | (2u << 30);                                    // type=2 (image)
    v8i g1;
    g1[0] = (int)(2u << 16);                               // data_size=4B
    g1[1] = (int)((dim0 & 0xFFFFu) << 16);                 // tensor_dim0[15:0]
    g1[2] = (int)(((dim0 >> 16) & 0xFFFFu) | (1u << 16));  // dim0[31:16] | tensor_dim1=1
    g1[3] = (int)((dim0 & 0xFFFFu) << 16);                 // tile_dim0 = nelem
    g1[4] = 1;                                             // tile_dim1 = 1
    g1[5] = (int)dim0;                                     // tensor_dim0_stride lo
    g1[6] = 0;
    g1[7] = 0;
    v4i z4 = {0, 0, 0, 0};
    v8i z8 = {0, 0, 0, 0, 0, 0, 0, 0};
    __builtin_amdgcn_tensor_load_to_lds(g0, g1, z4, z4, z8, 0);
}
#endif

// ------------------------------------------------- fused bilinear WMMA GEMM
// msg[e,o] = sum_{h,i} z[e,h] * x[src(e),i] * w2[h, i*H+o]  (+ x@b2, folded as
// an extra h-slab with zh=1), scatter-added into agg[dst(e), o].
// Block = 48 edges (3 M-tiles) x 128 threads; wave w owns output cols [16w,16w+16).
// w2 row (cin*H contiguous floats) double-buffered in LDS; TDM prefetches row
// h+1 while WMMAs consume row h (s_wait_tensorcnt pipelining).
template <int CIN>
__global__ __launch_bounds__(128)
void msg_wmma_kernel(const float* __restrict__ xin,        // [N, CIN]
                     const float* __restrict__ z,          // [E, H]
                     const float* __restrict__ w2,         // [H, CIN*H]
                     const float* __restrict__ b2,         // [CIN*H]
                     const int*   __restrict__ edge_index, // [2, E]
                     float*       __restrict__ agg)        // [N, H]
{
    constexpr int EPB  = 48;            // edges per block
    constexpr int MT   = EPB / 16;      // M-tiles
    constexpr int XS   = CIN + 1;       // padded x stride
    constexpr int KROW = CIN * H;       // one w2 row

    __shared__ float xs[EPB * XS];
    __shared__ float zs[EPB * 65];
    __shared__ float w2s[2][KROW];
    __shared__ int   dsts[EPB];

    const int tid  = threadIdx.x;
    const int wave = tid >> 5;
    const int lane = tid & 31;
    const int lh   = lane >> 4;         // 16x16x4 f32: lanes 0-15 hold K+0/1, 16-31 K+2/3
    const int col  = lane & 15;
    const int e0   = blockIdx.x * EPB;
    const int ocol = wave * 16 + col;

    // ---- stage gathered x rows, z rows, dst indices (tail-guarded)
    for (int idx = tid; idx < EPB * CIN; idx += 128) {
        int mm = idx / CIN, i = idx - mm * CIN;
        int e = e0 + mm; if (e >= NEDGES) e = NEDGES - 1;
        int src = edge_index[e];                           // row 0 = src
        xs[mm * XS + i] = xin[(size_t)src * CIN + i];
    }
    for (int idx = tid; idx < EPB * H; idx += 128) {
        int mm = idx >> 6, hh = idx & 63;
        int e = e0 + mm; if (e >= NEDGES) e = NEDGES - 1;
        zs[mm * 65 + hh] = z[(size_t)e * H + hh];
    }
    if (tid < EPB) {
        int e = e0 + tid; if (e >= NEDGES) e = NEDGES - 1;
        dsts[tid] = edge_index[NEDGES + e];                // row 1 = dst
    }
#if USE_TDM
    if (wave == 0)  // prologue: DMA row 0 into buffer 0
        tdm_load_row(w2, (unsigned)(size_t)&w2s[0][0], KROW);
#endif
    __syncthreads();

    v8f acc[MT];
#pragma unroll
    for (int mt = 0; mt < MT; ++mt)
        acc[mt] = (v8f){0.f, 0.f, 0.f, 0.f, 0.f, 0.f, 0.f, 0.f};

    for (int h = 0; h <= H; ++h) {      // h==H is the b2 slab (zh = 1)
        const int buf = h & 1;
#if USE_TDM
        if (wave == 0) {
            if (h < H) {                // prefetch next slab, then wait row h done
                const float* nrow = (h + 1 < H) ? (w2 + (size_t)(h + 1) * KROW) : b2;
                tdm_load_row(nrow, (unsigned)(size_t)&w2s[buf ^ 1][0], KROW);
                __builtin_amdgcn_s_wait_tensorcnt((short)1);
            } else {
                __builtin_amdgcn_s_wait_tensorcnt((short)0);
            }
        }
#else
        const float* rowp = (h < H) ? (w2 + (size_t)h * KROW) : b2;
        for (int idx = tid; idx < KROW; idx += 128) w2s[buf][idx] = rowp[idx];
        if (h + 1 < H && tid * 64 < KROW)
            __builtin_prefetch(w2 + (size_t)(h + 1) * KROW + tid * 64, 0, 1);
#endif
        __syncthreads();                // row h visible to all waves

        float zh[MT];
#pragma unroll
        for (int mt = 0; mt < MT; ++mt)
            zh[mt] = (h < H) ? zs[(mt * 16 + col) * 65 + h] : 1.0f;

#pragma unroll 4
        for (int i0 = 0; i0 < CIN; i0 += 4) {
            const int kk = i0 + 2 * lh;
            v2f b;                      // B fragment reused across all M-tiles
            b[0] = w2s[buf][kk * H + ocol];
            b[1] = w2s[buf][(kk + 1) * H + ocol];
#pragma unroll
            for (int mt = 0; mt < MT; ++mt) {
                const int mrow = mt * 16 + col;
                v2f a;
                a[0] = zh[mt] * xs[mrow * XS + kk];
                a[1] = zh[mt] * xs[mrow * XS + kk + 1];
                acc[mt] = __builtin_amdgcn_wmma_f32_16x16x4_f32(
                    false, a, false, b, (short)0, acc[mt], false, false);
            }
        }
        __syncthreads();                // buffer reuse fence
    }

    // ---- scatter-add; C/D layout: VGPR v <-> (M = v + 8*(lane>>4), N = lane&15)
#pragma unroll
    for (int mt = 0; mt < MT; ++mt) {
#pragma unroll
        for (int v = 0; v < 8; ++v) {
            int mv = mt * 16 + v + 8 * lh;
            int e  = e0 + mv;
            if (e < NEDGES)
                atomicAdd(&agg[(size_t)dsts[mv] * H + ocol], acc[mt][v]);
        }
    }
}

// ----------------------------------------- node update: +root, +bias, LN, relu
template <int CIN>
__global__ __launch_bounds__(64)
void node_update_kernel(const float* __restrict__ hin,   // [N, CIN]
                        const float* __restrict__ agg,   // [N, H]
                        const float* __restrict__ root,  // [CIN, H]
                        const float* __restrict__ bias,
                        const float* __restrict__ g,
                        const float* __restrict__ b,
                        float* __restrict__ hout)        // [N, H]
{
    __shared__ float s[H];
    __shared__ float stats[2];
    int n = blockIdx.x;
    int o = threadIdx.x;
    const float* hrow = hin + (size_t)n * CIN;
    float acc = agg[(size_t)n * H + o] + bias[o];
#pragma unroll 8
    for (int i = 0; i < CIN; ++i) acc += hrow[i] * root[i * H + o];
    s[o] = acc;
    __syncthreads();
    if (o == 0) {
        float mn = 0.f, vv = 0.f;
        for (int i = 0; i < H; ++i) mn += s[i];
        mn *= (1.0f / H);
        for (int i = 0; i < H; ++i) { float d = s[i] - mn; vv += d * d; }
        vv *= (1.0f / H);
        stats[0] = mn; stats[1] = vv;
    }
    __syncthreads();
    float y = (acc - stats[0]) * rsqrtf(stats[1] + 1e-5f) * g[o] + b[o];
    hout[(size_t)n * H + o] = fmaxf(y, 0.0f);
}

// ------------------------------------------------------- mean pool + head MLP
__global__ void pool_scatter_kernel(const float* __restrict__ h,
                                    const int* __restrict__ batch,
                                    float* __restrict__ hg,
                                    float* __restrict__ cnt) {
    int idx = blockIdx.x * blockDim.x + threadIdx.x;
    if (idx >= NNODES * H) return;
    int n = idx >> 6, o = idx & 63;
    int gi = batch[n];
    atomicAdd(&hg[(size_t)gi * H + o], h[idx]);
    if (o == 0) atomicAdd(&cnt[gi], 1.0f);
}

__global__ __launch_bounds__(32)
void head_kernel(const float* __restrict__ hg, const float* __restrict__ cnt,
                 const float* __restrict__ fc1w, const float* __restrict__ fc1b,
                 const float* __restrict__ fc2w, const float* __restrict__ fc2b,
                 float* __restrict__ out) {
    __shared__ float t[32];
    int gi = blockIdx.x;
    int j = threadIdx.x;
    float c = fmaxf(cnt[gi], 1.0f);
    float acc = fc1b[j];
#pragma unroll 8
    for (int h = 0; h < H; ++h) acc += (hg[(size_t)gi * H + h] / c) * fc1w[h * 32 + j];
    t[j] = fmaxf(acc, 0.0f);
    __syncthreads();
    if (j == 0) {
        float r = fc2b[0];
        for (int k = 0; k < 32; ++k) r += t[k] * fc2w[k];
        out[gi] = r;
    }
}

// ----------------------------------------------------------------- launcher
extern "C" void kernel_launch(void* const* d_in, const int* in_sizes, int n_in,
                              void* d_out, int out_size, void* d_ws, size_t ws_size,
                              hipStream_t stream) {
    const float* x     = (const float*)d_in[0];
    const int*   ei    = (const int*)d_in[1];
    const float* ea    = (const float*)d_in[2];
    const int*   batch = (const int*)d_in[3];
    auto P = [&](int i) { return (const float*)d_in[i]; };

    float* ws  = (float*)d_ws;
    float* hA  = ws;                                   // [N,H]
    float* hB  = hA  + (size_t)NNODES * H;             // [N,H]
    float* zb  = hB  + (size_t)NNODES * H;             // [E,H]
    float* agg = zb  + (size_t)NEDGES * H;             // [N,H]
    float* hg  = agg + (size_t)NNODES * H;             // [G,H]
    float* cnt = hg  + (size_t)NG * H;                 // [G]
    float* out = (float*)d_out;

    const int NH = NNODES * H, EH = NEDGES * H;
    const int TPB = 256;
    const int MSG_BLOCKS = (NEDGES / 16 + 2) / 3;      // 48 edges per block

    // ---------------- layer 1 (cin = 32, input = x)
    zero_kernel<<<(NH + TPB - 1) / TPB, TPB, 0, stream>>>(agg, NH);
    edge_mlp1_kernel<<<(EH + TPB - 1) / TPB, TPB, 0, stream>>>(ea, P(4), P(5), zb);
    msg_wmma_kernel<32><<<MSG_BLOCKS, 128, 0, stream>>>(x, zb, P(6), P(7), ei, agg);
    node_update_kernel<32><<<NNODES, 64, 0, stream>>>(x, agg, P(8), P(9), P(10), P(11), hA);

    // ---------------- layer 2 (cin = 64)
    zero_kernel<<<(NH + TPB - 1) / TPB, TPB, 0, stream>>>(agg, NH);
    edge_mlp1_kernel<<<(EH + TPB - 1) / TPB, TPB, 0, stream>>>(ea, P(12), P(13), zb);
    msg_wmma_kernel<64><<<MSG_BLOCKS, 128, 0, stream>>>(hA, zb, P(14), P(15), ei, agg);
    node_update_kernel<64><<<NNODES, 64, 0, stream>>>(hA, agg, P(16), P(17), P(18), P(19), hB);

    // ---------------- layer 3 (cin = 64)
    zero_kernel<<<(NH + TPB - 1) / TPB, TPB, 0, stream>>>(agg, NH);
    edge_mlp1_kernel<<<(EH + TPB - 1) / TPB, TPB, 0, stream>>>(ea, P(20), P(21), zb);
    msg_wmma_kernel<64><<<MSG_BLOCKS, 128, 0, stream>>>(hB, zb, P(22), P(23), ei, agg);
    node_update_kernel<64><<<NNODES, 64, 0, stream>>>(hB, agg, P(24), P(25), P(26), P(27), hA);

    // ---------------- global mean pool + head
    zero_kernel<<<(NG * H + NG + TPB - 1) / TPB, TPB, 0, stream>>>(hg, NG * H + NG);
    pool_scatter_kernel<<<(NH + TPB - 1) / TPB, TPB, 0, stream>>>(hA, batch, hg, cnt);
    head_kernel<<<NG, 32, 0, stream>>>(hg, cnt, P(28), P(29), P(30), P(31), out);
}